// GramEmbedding_63694365000012
// MI455X (gfx1250) — compile-verified
//
#include <hip/hip_runtime.h>
#include <math.h>

#define BB 4096
#define VV 2048
#define EE 16
#define DD 16
#define HH 16
#define KCHUNK 256
#define SPLITK 8

typedef __attribute__((ext_vector_type(2))) float v2f;
typedef __attribute__((ext_vector_type(4))) float v4f;
typedef __attribute__((ext_vector_type(8))) float v8f;
typedef __attribute__((ext_vector_type(4))) int   v4i;

typedef __attribute__((address_space(1))) v4i g_v4i;   // global
typedef __attribute__((address_space(3))) v4i l_v4i;   // LDS

#if defined(__has_builtin)
#if __has_builtin(__builtin_amdgcn_global_load_async_to_lds_b128)
#define HAVE_ASYNC_LDS 1
#endif
#endif

// ---------------------------------------------------------------------------
// Kernel 1: score[v,e] = tanh( emb[v]·W1 + anc[v,e]·W2 + b1 + b2 ) · u + u_b
// ---------------------------------------------------------------------------
__global__ __launch_bounds__(256) void score_kernel(
    const float* __restrict__ conc_emb,   // (V,1,D)
    const float* __restrict__ conc_anc,   // (V,E,D)
    const float* __restrict__ w1_w,       // (D,H)
    const float* __restrict__ w1_b,       // (H)
    const float* __restrict__ w2_w,       // (D,H)
    const float* __restrict__ w2_b,       // (H)
    const float* __restrict__ u_w,        // (H,1)
    const float* __restrict__ u_b,        // (1)
    float* __restrict__ score)            // (V,E)
{
    __shared__ float s_w1[DD * HH];
    __shared__ float s_w2[DD * HH];
    __shared__ float s_u[HH];
    __shared__ float s_b1[HH];
    __shared__ float s_b2[HH];

    int tid = threadIdx.x;
    s_w1[tid] = w1_w[tid];
    s_w2[tid] = w2_w[tid];
    if (tid < HH) {
        s_u[tid]  = u_w[tid];
        s_b1[tid] = w1_b[tid];
        s_b2[tid] = w2_b[tid];
    }
    __syncthreads();

    int idx = blockIdx.x * 256 + tid;     // 0 .. V*E-1
    int v = idx >> 4;
    int e = idx & 15;

    float emb[DD], anc[DD];
#pragma unroll
    for (int d = 0; d < DD; ++d) {
        emb[d] = conc_emb[v * DD + d];
        anc[d] = conc_anc[(v * EE + e) * DD + d];
    }

    float sc = 0.0f;
#pragma unroll
    for (int h = 0; h < HH; ++h) {
        float h1 = s_b1[h];
        float h2 = s_b2[h];
#pragma unroll
        for (int d = 0; d < DD; ++d) {
            h1 += emb[d] * s_w1[d * HH + h];
            h2 += anc[d] * s_w2[d * HH + h];
        }
        sc += tanhf(h1 + h2) * s_u[h];
    }
    score[idx] = sc + u_b[0];
}

// ---------------------------------------------------------------------------
// Kernel 2: per-e softmax stats over the V axis. blockIdx.x = e (16 blocks).
// ---------------------------------------------------------------------------
__global__ __launch_bounds__(256) void stats_kernel(
    const float* __restrict__ score, float* __restrict__ stats)
{
    __shared__ float red[256];
    int e = blockIdx.x;
    int tid = threadIdx.x;

    float m = -INFINITY;
    for (int v = tid; v < VV; v += 256)
        m = fmaxf(m, score[v * EE + e]);
    red[tid] = m;
    __syncthreads();
#pragma unroll
    for (int s = 128; s > 0; s >>= 1) {
        if (tid < s) red[tid] = fmaxf(red[tid], red[tid + s]);
        __syncthreads();
    }
    m = red[0];
    __syncthreads();

    float sum = 0.0f;
    for (int v = tid; v < VV; v += 256)
        sum += expf(score[v * EE + e] - m);
    red[tid] = sum;
    __syncthreads();
#pragma unroll
    for (int s = 128; s > 0; s >>= 1) {
        if (tid < s) red[tid] += red[tid + s];
        __syncthreads();
    }
    if (tid == 0) {
        stats[2 * e]     = m;
        stats[2 * e + 1] = red[0];
    }
}

// ---------------------------------------------------------------------------
// Kernel 3: context[v,d] = sum_e softmax(score)[v,e] * conc_anc[v,e,d]
// ---------------------------------------------------------------------------
__global__ __launch_bounds__(256) void context_kernel(
    const float* __restrict__ score,
    const float* __restrict__ stats,
    const float* __restrict__ conc_anc,
    float* __restrict__ context)          // (V, D) row-major
{
    int idx = blockIdx.x * 256 + threadIdx.x;   // 0 .. V*D-1
    int v = idx >> 4;
    int d = idx & 15;

    float acc = 0.0f;
#pragma unroll
    for (int e = 0; e < EE; ++e) {
        float attn = expf(score[v * EE + e] - stats[2 * e]) / stats[2 * e + 1];
        acc += attn * conc_anc[(v * EE + e) * DD + d];
    }
    context[idx] = acc;
}

// ---------------------------------------------------------------------------
// Kernel 4: out_partial[split] = values(4096x2048) @ context(2048x16) chunk
// via V_WMMA_F32_16X16X4_F32.  Block = 8 waves sharing one K-chunk of context
// staged in LDS (async-to-LDS path); each wave streams a 16-row strip of
// `values` straight into A fragments.
// ---------------------------------------------------------------------------
__global__ __launch_bounds__(256) void gemm_kernel(
    const float* __restrict__ values,     // (B, V)
    const float* __restrict__ context,    // (V, D)
    float* __restrict__ partial)          // (SPLITK, B, D)
{
    __shared__ float sB[KCHUNK * DD];     // 16 KB context slice

    int tg    = blockIdx.x >> 3;          // 0..31 : group of 8 row-tiles
    int split = blockIdx.x & 7;           // 0..7  : K chunk
    int k0    = split * KCHUNK;

    // stage context[k0 .. k0+255][0..15] into LDS
    {
        const v4f* src = reinterpret_cast<const v4f*>(context + k0 * DD);
        v4f* dst = reinterpret_cast<v4f*>(sB);
#ifdef HAVE_ASYNC_LDS
        for (int i = threadIdx.x; i < (KCHUNK * DD) / 4; i += 256) {
            __builtin_amdgcn_global_load_async_to_lds_b128(
                (g_v4i*)(src + i), (l_v4i*)(dst + i), 0, 0);
        }
#if __has_builtin(__builtin_amdgcn_s_wait_asynccnt)
        __builtin_amdgcn_s_wait_asynccnt(0);
#else
        asm volatile("s_wait_asynccnt 0x0" ::: "memory");
#endif
#else
        for (int i = threadIdx.x; i < (KCHUNK * DD) / 4; i += 256)
            dst[i] = src[i];
#endif
    }
    __syncthreads();

    int wave = threadIdx.x >> 5;          // 0..7
    int lane = threadIdx.x & 31;
    int col  = lane & 15;
    int hi   = lane >> 4;                 // 0 or 1 (half-wave)

    int rowBase = tg * 128 + wave * 16;
    int row     = rowBase + col;          // M = lane%16 for A fragment

    const v2f* ap =
        reinterpret_cast<const v2f*>(values + (size_t)row * VV + k0);

    v8f acc = {};
#pragma unroll 8
    for (int k = 0; k < KCHUNK; k += 4) {
        v2f a = ap[(k >> 1) + hi];        // {K=k+2hi, K=k+1+2hi} for row M
        v2f b;
        b.x = sB[(k + 2 * hi) * DD + col];
        b.y = sB[(k + 1 + 2 * hi) * DD + col];
        acc = __builtin_amdgcn_wmma_f32_16x16x4_f32(
            false, a, false, b, (short)0, acc, false, false);
    }

    float* dst = partial + (size_t)split * (BB * DD);
#pragma unroll
    for (int r = 0; r < 8; ++r) {
        int orow = rowBase + r + 8 * hi;
        dst[orow * DD + col] = acc[r];
    }
}

// ---------------------------------------------------------------------------
// Kernel 5: out = sum over splits of partial (float4 per thread)
// ---------------------------------------------------------------------------
__global__ __launch_bounds__(256) void reduce_kernel(
    const float* __restrict__ partial, float* __restrict__ out)
{
    int idx = blockIdx.x * 256 + threadIdx.x;   // 0 .. B*D/4-1
    const v4f* p = reinterpret_cast<const v4f*>(partial);
    v4f acc = {};
#pragma unroll
    for (int s = 0; s < SPLITK; ++s) {
        v4f t = p[(size_t)s * (BB * DD / 4) + idx];
        acc.x += t.x; acc.y += t.y; acc.z += t.z; acc.w += t.w;
    }
    reinterpret_cast<v4f*>(out)[idx] = acc;
}

// ---------------------------------------------------------------------------
extern "C" void kernel_launch(void* const* d_in, const int* in_sizes, int n_in,
                              void* d_out, int out_size, void* d_ws, size_t ws_size,
                              hipStream_t stream) {
    const float* values   = (const float*)d_in[0];
    const float* conc_emb = (const float*)d_in[1];
    const float* conc_anc = (const float*)d_in[2];
    const float* w1_w     = (const float*)d_in[3];
    const float* w1_b     = (const float*)d_in[4];
    const float* w2_w     = (const float*)d_in[5];
    const float* w2_b     = (const float*)d_in[6];
    const float* u_w      = (const float*)d_in[7];
    const float* u_b      = (const float*)d_in[8];
    float* out = (float*)d_out;

    float* ws      = (float*)d_ws;
    float* score   = ws;                       // V*E        = 32768
    float* stats   = ws + 32768;               // 32
    float* context = ws + 32800;               // V*D        = 32768
    float* partial = ws + 65568;               // SPLITK*B*D = 524288

    score_kernel<<<(VV * EE) / 256, 256, 0, stream>>>(
        conc_emb, conc_anc, w1_w, w1_b, w2_w, w2_b, u_w, u_b, score);
    stats_kernel<<<EE, 256, 0, stream>>>(score, stats);
    context_kernel<<<(VV * DD) / 256, 256, 0, stream>>>(
        score, stats, conc_anc, context);
    gemm_kernel<<<32 * SPLITK, 256, 0, stream>>>(values, context, partial);
    reduce_kernel<<<(BB * DD) / 1024, 256, 0, stream>>>(partial, out);
}